// GNNNet_51634096833177
// MI455X (gfx1250) — compile-verified
//
#include <hip/hip_runtime.h>
#include <hip/hip_bf16.h>

typedef __attribute__((ext_vector_type(2))) float v2f;
typedef __attribute__((ext_vector_type(8))) float v8f;

#define NEG_SLOPE 0.2f

__device__ __forceinline__ float lrelu(float v) { return v > 0.f ? v : NEG_SLOPE * v; }

// float atomic-max via sign-split int/uint atomics (correct with -inf init)
__device__ __forceinline__ void atomicMaxF(float* addr, float val) {
  if (val >= 0.f) atomicMax((int*)addr, __float_as_int(val));
  else            atomicMin((unsigned int*)addr, __float_as_uint(val));
}

// ---------------- Layer 1: h1 = x * W1 row, alpha_s/d, init m/s/accum -------
// one wave (32 lanes) per node; lane owns channels hc = lane + 32k (head k)
__global__ void k_layer1(const float* __restrict__ x, const float* __restrict__ W1,
                         const float* __restrict__ asrc, const float* __restrict__ adst,
                         float* __restrict__ h, float* __restrict__ alpha_s,
                         float* __restrict__ alpha_d, float* __restrict__ m,
                         float* __restrict__ s, float* __restrict__ accum, int N) {
  int wid  = (blockIdx.x * blockDim.x + threadIdx.x) >> 5;
  int lane = threadIdx.x & 31;
  if (wid >= N) return;
  float xv = x[wid];
  float ps[4], pd[4];
#pragma unroll
  for (int k = 0; k < 4; ++k) {
    int hc = lane + (k << 5);
    float hv = xv * W1[hc];
    h[(size_t)wid * 128 + hc] = hv;
    accum[(size_t)wid * 128 + hc] = 0.f;
    ps[k] = hv * asrc[hc];
    pd[k] = hv * adst[hc];
  }
#pragma unroll
  for (int off = 16; off; off >>= 1) {
#pragma unroll
    for (int k = 0; k < 4; ++k) {
      ps[k] += __shfl_down(ps[k], off, 32);
      pd[k] += __shfl_down(pd[k], off, 32);
    }
  }
  if (lane == 0) {
#pragma unroll
    for (int k = 0; k < 4; ++k) {
      alpha_s[(size_t)wid * 4 + k] = ps[k];
      alpha_d[(size_t)wid * 4 + k] = pd[k];
      m[(size_t)wid * 4 + k] = __int_as_float(0xff800000); // -inf
      s[(size_t)wid * 4 + k] = 0.f;
    }
  }
}

// ------------- per-node alpha from precomputed h (layer 2), + inits --------
__global__ void k_node_alpha(const float* __restrict__ h, const float* __restrict__ asrc,
                             const float* __restrict__ adst, float* __restrict__ alpha_s,
                             float* __restrict__ alpha_d, float* __restrict__ m,
                             float* __restrict__ s, float* __restrict__ accum, int N) {
  int wid  = (blockIdx.x * blockDim.x + threadIdx.x) >> 5;
  int lane = threadIdx.x & 31;
  if (wid >= N) return;
  float ps[4], pd[4];
#pragma unroll
  for (int k = 0; k < 4; ++k) {
    int hc = lane + (k << 5);
    float hv = h[(size_t)wid * 128 + hc];
    accum[(size_t)wid * 128 + hc] = 0.f;
    ps[k] = hv * asrc[hc];
    pd[k] = hv * adst[hc];
  }
#pragma unroll
  for (int off = 16; off; off >>= 1) {
#pragma unroll
    for (int k = 0; k < 4; ++k) {
      ps[k] += __shfl_down(ps[k], off, 32);
      pd[k] += __shfl_down(pd[k], off, 32);
    }
  }
  if (lane == 0) {
#pragma unroll
    for (int k = 0; k < 4; ++k) {
      alpha_s[(size_t)wid * 4 + k] = ps[k];
      alpha_d[(size_t)wid * 4 + k] = pd[k];
      m[(size_t)wid * 4 + k] = __int_as_float(0xff800000);
      s[(size_t)wid * 4 + k] = 0.f;
    }
  }
}

// ---------------- edge pass A: segment max (H=4) ---------------------------
__global__ void k_edge_max4(const int* __restrict__ src, const int* __restrict__ dst,
                            const float* __restrict__ as_, const float* __restrict__ ad_,
                            float* __restrict__ m, int E, int Etot) {
  int e = blockIdx.x * blockDim.x + threadIdx.x;
  if (e >= Etot) return;
  int sN, dN;
  if (e < E) { sN = src[e]; dN = dst[e]; } else { sN = dN = e - E; }
  float4 es = *(const float4*)(as_ + 4 * (size_t)sN);
  float4 ed = *(const float4*)(ad_ + 4 * (size_t)dN);
  atomicMaxF(&m[4 * (size_t)dN + 0], lrelu(es.x + ed.x));
  atomicMaxF(&m[4 * (size_t)dN + 1], lrelu(es.y + ed.y));
  atomicMaxF(&m[4 * (size_t)dN + 2], lrelu(es.z + ed.z));
  atomicMaxF(&m[4 * (size_t)dN + 3], lrelu(es.w + ed.w));
}

// ---- edge pass B: fused exp + denom accumulate + unnormalized aggregation -
// one wave per edge; lane owns channels hc = lane + 32k (head k)
__global__ void k_edge_aggr4(const int* __restrict__ src, const int* __restrict__ dst,
                             const float* __restrict__ as_, const float* __restrict__ ad_,
                             const float* __restrict__ m, float* __restrict__ s,
                             const float* __restrict__ h, float* __restrict__ accum,
                             int E, int Etot) {
  int e    = (blockIdx.x * blockDim.x + threadIdx.x) >> 5;
  int lane = threadIdx.x & 31;
  if (e >= Etot) return;
  int sN, dN;
  if (e < E) { sN = src[e]; dN = dst[e]; } else { sN = dN = e - E; }
  float4 es = *(const float4*)(as_ + 4 * (size_t)sN);
  float4 ed = *(const float4*)(ad_ + 4 * (size_t)dN);
  float4 mm = *(const float4*)(m + 4 * (size_t)dN);
  float p0 = __expf(lrelu(es.x + ed.x) - mm.x);
  float p1 = __expf(lrelu(es.y + ed.y) - mm.y);
  float p2 = __expf(lrelu(es.z + ed.z) - mm.z);
  float p3 = __expf(lrelu(es.w + ed.w) - mm.w);
  if (lane < 4) {
    float pl = (lane == 0) ? p0 : (lane == 1) ? p1 : (lane == 2) ? p2 : p3;
    atomicAdd(&s[4 * (size_t)dN + lane], pl);
  }
  const float* hs = h + (size_t)sN * 128;
  float* od = accum + (size_t)dN * 128;
  atomicAdd(&od[lane +  0], p0 * hs[lane +  0]);
  atomicAdd(&od[lane + 32], p1 * hs[lane + 32]);
  atomicAdd(&od[lane + 64], p2 * hs[lane + 64]);
  atomicAdd(&od[lane + 96], p3 * hs[lane + 96]);
}

// ------------- normalize + bias + ELU  (a = elu(accum/s + b)) --------------
__global__ void k_finalize4(const float* __restrict__ accum, const float* __restrict__ s,
                            const float* __restrict__ b, float* __restrict__ out, int N) {
  int i = blockIdx.x * blockDim.x + threadIdx.x;
  if (i >= N * 128) return;
  int hc = i & 127;
  int hd = hc >> 5;
  int n  = i >> 7;
  float v = accum[i] / s[(size_t)n * 4 + hd] + b[hc];
  out[i] = v > 0.f ? v : __expf(v) - 1.f;
}

// --------------- Layer-2 GEMM: D[N,128] = A[N,128] x B[128,128] ------------
// block = 256 threads = 8 waves; block owns one 16-row M tile; wave w owns
// N-tile w. K-loop of 32 x V_WMMA_F32_16X16X4_F32.
__global__ void k_gemm128(const float* __restrict__ A, const float* __restrict__ Bm,
                          float* __restrict__ D, int Nrows) {
  __shared__ float sA[16 * 130];   // padded stride: stride%64 != 0, even words
  const int t     = threadIdx.x;
  const int mtile = blockIdx.x;
  for (int i = t; i < 2048; i += 256) {
    int r = i >> 7, c = i & 127;
    int grow = mtile * 16 + r;
    sA[r * 130 + c] = (grow < Nrows) ? A[(size_t)grow * 128 + c] : 0.f;
  }
  __syncthreads();
  const int wave = t >> 5;
  const int lane = t & 31;
  const int r    = lane & 15;        // row (A) / col (B,D) within tile
  const int kh   = (lane >> 4) << 1; // lane-half selects K pair {0,1} vs {2,3}
  v8f acc = {};
  const float* Bbase = Bm + wave * 16 + r;
#pragma unroll
  for (int k = 0; k < 32; ++k) {
    const int kk = (k << 2) + kh;
    v2f a, b;
    a[0] = sA[r * 130 + kk];
    a[1] = sA[r * 130 + kk + 1];
    b[0] = Bbase[(size_t)kk * 128];
    b[1] = Bbase[(size_t)(kk + 1) * 128];
    acc = __builtin_amdgcn_wmma_f32_16x16x4_f32(false, a, false, b,
                                                (short)0, acc, false, false);
  }
  const int moff = (lane >> 4) * 8;  // D: vgpr j -> row j + 8*(lane/16)
#pragma unroll
  for (int j = 0; j < 8; ++j) {
    int row = mtile * 16 + moff + j;
    if (row < Nrows) D[(size_t)row * 128 + wave * 16 + r] = acc[j];
  }
}

// ---------------- Layer 3 (H=1, C=1) ---------------------------------------
__global__ void k_layer3_node(const float* __restrict__ a2, const float* __restrict__ W3,
                              const float* __restrict__ asrc, const float* __restrict__ adst,
                              float* __restrict__ h3, float* __restrict__ as3,
                              float* __restrict__ ad3, float* __restrict__ m3,
                              float* __restrict__ s3, float* __restrict__ acc3, int N) {
  int wid  = (blockIdx.x * blockDim.x + threadIdx.x) >> 5;
  int lane = threadIdx.x & 31;
  if (wid >= N) return;
  float part = 0.f;
#pragma unroll
  for (int k = 0; k < 4; ++k) {
    int c = lane + (k << 5);
    part += a2[(size_t)wid * 128 + c] * W3[c];
  }
#pragma unroll
  for (int off = 16; off; off >>= 1) part += __shfl_down(part, off, 32);
  if (lane == 0) {
    h3[wid]  = part;
    as3[wid] = part * asrc[0];
    ad3[wid] = part * adst[0];
    m3[wid]  = __int_as_float(0xff800000);
    s3[wid]  = 0.f;
    acc3[wid] = 0.f;
  }
}

__global__ void k_edge_max1(const int* __restrict__ src, const int* __restrict__ dst,
                            const float* __restrict__ as_, const float* __restrict__ ad_,
                            float* __restrict__ m, int E, int Etot) {
  int e = blockIdx.x * blockDim.x + threadIdx.x;
  if (e >= Etot) return;
  int sN, dN;
  if (e < E) { sN = src[e]; dN = dst[e]; } else { sN = dN = e - E; }
  atomicMaxF(&m[dN], lrelu(as_[sN] + ad_[dN]));
}

__global__ void k_edge_aggr1(const int* __restrict__ src, const int* __restrict__ dst,
                             const float* __restrict__ as_, const float* __restrict__ ad_,
                             const float* __restrict__ m, float* __restrict__ s,
                             const float* __restrict__ h3, float* __restrict__ acc3,
                             int E, int Etot) {
  int e = blockIdx.x * blockDim.x + threadIdx.x;
  if (e >= Etot) return;
  int sN, dN;
  if (e < E) { sN = src[e]; dN = dst[e]; } else { sN = dN = e - E; }
  float p = __expf(lrelu(as_[sN] + ad_[dN]) - m[dN]);
  atomicAdd(&s[dN], p);
  atomicAdd(&acc3[dN], p * h3[sN]);
}

__global__ void k_out3(const float* __restrict__ acc3, const float* __restrict__ s3,
                       const float* __restrict__ b3, float* __restrict__ out, int N) {
  int n = blockIdx.x * blockDim.x + threadIdx.x;
  if (n >= N) return;
  out[n] = acc3[n] / s3[n] + b3[0];
}

extern "C" void kernel_launch(void* const* d_in, const int* in_sizes, int n_in,
                              void* d_out, int out_size, void* d_ws, size_t ws_size,
                              hipStream_t stream) {
  const float* x    = (const float*)d_in[0];
  const int*   ei   = (const int*)d_in[1];
  const float* W1   = (const float*)d_in[2];
  const float* as1w = (const float*)d_in[3];
  const float* ad1w = (const float*)d_in[4];
  const float* b1   = (const float*)d_in[5];
  const float* W2   = (const float*)d_in[6];
  const float* as2w = (const float*)d_in[7];
  const float* ad2w = (const float*)d_in[8];
  const float* b2   = (const float*)d_in[9];
  const float* W3   = (const float*)d_in[10];
  const float* as3w = (const float*)d_in[11];
  const float* ad3w = (const float*)d_in[12];
  const float* b3   = (const float*)d_in[13];

  const int N = in_sizes[0];          // x is [N,1]
  const int E = in_sizes[1] / 2;      // edge_index is [2,E]
  const int Etot = E + N;             // + self-loops
  const int* srcIdx = ei;
  const int* dstIdx = ei + (size_t)E;

  float* ws = (float*)d_ws;
  size_t o = 0;
  float* h1    = ws + o; o += (size_t)N * 128;  // h1, later a1, later layer-3 scratch
  float* h2    = ws + o; o += (size_t)N * 128;  // h2, later a2
  float* accum = ws + o; o += (size_t)N * 128;
  float* as_   = ws + o; o += (size_t)N * 4;
  float* ad_   = ws + o; o += (size_t)N * 4;
  float* m_    = ws + o; o += (size_t)N * 4;
  float* s_    = ws + o; o += (size_t)N * 4;
  // layer-3 per-node scalars alias the (no longer needed) h1 region
  float* h3   = h1;
  float* as3  = h1 + (size_t)N;
  float* ad3  = h1 + (size_t)2 * N;
  float* m3   = h1 + (size_t)3 * N;
  float* s3   = h1 + (size_t)4 * N;
  float* acc3 = h1 + (size_t)5 * N;

  const int nodeWaveBlocks = (N + 7) / 8;          // 8 waves/block
  const int edgeBlocks     = (Etot + 255) / 256;
  const int edgeWaveBlocks = (Etot + 7) / 8;
  const int featBlocks     = (N * 128 + 255) / 256;
  const int mTiles         = (N + 15) / 16;

  // ---- Layer 1 ----
  k_layer1<<<nodeWaveBlocks, 256, 0, stream>>>(x, W1, as1w, ad1w, h1, as_, ad_, m_, s_, accum, N);
  k_edge_max4<<<edgeBlocks, 256, 0, stream>>>(srcIdx, dstIdx, as_, ad_, m_, E, Etot);
  k_edge_aggr4<<<edgeWaveBlocks, 256, 0, stream>>>(srcIdx, dstIdx, as_, ad_, m_, s_, h1, accum, E, Etot);
  k_finalize4<<<featBlocks, 256, 0, stream>>>(accum, s_, b1, h1, N);   // a1 -> h1 buffer

  // ---- Layer 2 ----
  k_gemm128<<<mTiles, 256, 0, stream>>>(h1, W2, h2, N);
  k_node_alpha<<<nodeWaveBlocks, 256, 0, stream>>>(h2, as2w, ad2w, as_, ad_, m_, s_, accum, N);
  k_edge_max4<<<edgeBlocks, 256, 0, stream>>>(srcIdx, dstIdx, as_, ad_, m_, E, Etot);
  k_edge_aggr4<<<edgeWaveBlocks, 256, 0, stream>>>(srcIdx, dstIdx, as_, ad_, m_, s_, h2, accum, E, Etot);
  k_finalize4<<<featBlocks, 256, 0, stream>>>(accum, s_, b2, h2, N);   // a2 -> h2 buffer

  // ---- Layer 3 ----
  k_layer3_node<<<nodeWaveBlocks, 256, 0, stream>>>(h2, W3, as3w, ad3w, h3, as3, ad3, m3, s3, acc3, N);
  k_edge_max1<<<edgeBlocks, 256, 0, stream>>>(srcIdx, dstIdx, as3, ad3, m3, E, Etot);
  k_edge_aggr1<<<edgeBlocks, 256, 0, stream>>>(srcIdx, dstIdx, as3, ad3, m3, s3, h3, acc3, E, Etot);
  k_out3<<<(N + 255) / 256, 256, 0, stream>>>(acc3, s3, b3, (float*)d_out, N);
}